// StochasticMultiLayerRGCN_56968446214863
// MI455X (gfx1250) — compile-verified
//
#include <hip/hip_runtime.h>
#include <hip/hip_bf16.h>

typedef __attribute__((ext_vector_type(2))) float v2f;
typedef __attribute__((ext_vector_type(8))) float v8f;

#define H_HEADS 4
#define DHEAD   64
#define NCOLS   256   // H*D

// ---------------------------------------------------------------------------
// WMMA GEMM: Z[n x 256] = X[n x K] @ W[K x 256], fp32, V_WMMA_F32_16X16X4_F32
// block = 128 threads (4 waves), each wave owns one 16x16 output tile.
// grid.x covers M tiles / 4, grid.y = 256/16 = 16 N tiles.
// ---------------------------------------------------------------------------
__global__ void gemm_wmma_f32(const float* __restrict__ X,
                              const float* __restrict__ W,
                              float* __restrict__ Z,
                              int n, int K) {
    const int wave = threadIdx.x >> 5;
    const int lane = threadIdx.x & 31;
    const int tm   = blockIdx.x * 4 + wave;     // M tile index
    const int m0   = tm * 16;
    if (m0 >= n) return;                        // wave-uniform exit
    const int n0   = blockIdx.y * 16;
    const int half = lane >> 4;                 // 0: lanes 0-15, 1: lanes 16-31
    const int l16  = lane & 15;

    // A-row for this lane (clamped + masked so EXEC stays all-ones)
    const int   m    = m0 + l16;
    const int   mc   = (m < n) ? m : (n - 1);
    const float mask = (m < n) ? 1.0f : 0.0f;
    const float* __restrict__ xrow = X + (size_t)mc * K;

    v8f acc = {};
    for (int k0 = 0; k0 < K; k0 += 4) {
        const int ka = k0 + half * 2;           // K index of vgpr0 element
        v2f a, b;
        a[0] = xrow[ka + 0] * mask;             // A 16x4: lane=M, vgpr->K (+2 for upper half)
        a[1] = xrow[ka + 1] * mask;
        b[0] = W[(size_t)(ka + 0) * NCOLS + n0 + l16];  // B 4x16: lane=N, vgpr->K
        b[1] = W[(size_t)(ka + 1) * NCOLS + n0 + l16];
        acc = __builtin_amdgcn_wmma_f32_16x16x4_f32(
            /*neg_a=*/false, a, /*neg_b=*/false, b,
            /*c_mod=*/(short)0, acc, /*reuse_a=*/false, /*reuse_b=*/false);
    }
    // C/D layout: vgpr r, lanes 0-15 -> M=r, lanes 16-31 -> M=r+8, N = lane&15
    #pragma unroll
    for (int r = 0; r < 8; ++r) {
        const int mr = m0 + r + half * 8;
        if (mr < n) Z[(size_t)mr * NCOLS + n0 + l16] = acc[r];
    }
}

// el[i,h] = dot(z[i,h,:], attn[h,:])   (one thread per (node,head))
__global__ void head_dot(const float* __restrict__ Z, const float* __restrict__ attn,
                         float* __restrict__ out, int n) {
    int i = blockIdx.x * blockDim.x + threadIdx.x;
    if (i >= n * H_HEADS) return;
    const int node = i >> 2, h = i & 3;
    const float* z = Z + (size_t)node * NCOLS + h * DHEAD;
    const float* a = attn + h * DHEAD;
    float s = 0.f;
    #pragma unroll 8
    for (int d = 0; d < DHEAD; ++d) s += z[d] * a[d];
    out[i] = s;
}

__global__ void fill_u32(unsigned* p, unsigned v, int n) {
    int i = blockIdx.x * blockDim.x + threadIdx.x;
    if (i < n) p[i] = v;
}
__global__ void fill_f32(float* p, float v, int n) {
    int i = blockIdx.x * blockDim.x + threadIdx.x;
    if (i < n) p[i] = v;
}

__device__ __forceinline__ unsigned enc_ord(float f) {
    unsigned u = __float_as_uint(f);
    return (u & 0x80000000u) ? ~u : (u | 0x80000000u);
}
__device__ __forceinline__ float dec_ord(unsigned u) {
    return __uint_as_float((u & 0x80000000u) ? (u ^ 0x80000000u) : ~u);
}

// pass 1: segment max of leaky-relu'd logits (order-preserving u32 atomicMax)
__global__ void edge_max(const int* __restrict__ src, const int* __restrict__ dst,
                         const float* __restrict__ el, const float* __restrict__ er,
                         unsigned* __restrict__ emax, int E) {
    int i = blockIdx.x * blockDim.x + threadIdx.x;
    if (i >= E) return;
    const int s = src[i], d = dst[i];
    #pragma unroll
    for (int h = 0; h < H_HEADS; ++h) {
        float e = el[s * H_HEADS + h] + er[d * H_HEADS + h];
        e = (e > 0.f) ? e : 0.2f * e;
        atomicMax(&emax[d * H_HEADS + h], enc_ord(e));
    }
}

// pass 2: ex = exp(e - emax[dst]); denom[dst,h] += ex; unnorm[dst] += ex * z_src[src]
// one wave per edge, each lane scatters 8 of the 256 features (all in one head)
__global__ void edge_scatter(const int* __restrict__ src, const int* __restrict__ dst,
                             const float* __restrict__ el, const float* __restrict__ er,
                             const unsigned* __restrict__ emax,
                             const float* __restrict__ zsrc,
                             float* __restrict__ denom, float* __restrict__ unnorm, int E) {
    const int e_id = (blockIdx.x * blockDim.x + threadIdx.x) >> 5;
    const int lane = threadIdx.x & 31;
    if (e_id >= E) return;
    const int s = src[e_id], d = dst[e_id];
    const int h = lane >> 3;                    // 8 lanes per head
    float e = el[s * H_HEADS + h] + er[d * H_HEADS + h];
    e = (e > 0.f) ? e : 0.2f * e;
    float m = dec_ord(emax[d * H_HEADS + h]);
    if (!isfinite(m)) m = 0.f;                  // reference: where(isfinite(emax), emax, 0)
    const float ex = __expf(e - m);
    if ((lane & 7) == 0) atomicAdd(&denom[d * H_HEADS + h], ex);
    const float* __restrict__ z = zsrc + (size_t)s * NCOLS;
    float* __restrict__ o = unnorm + (size_t)d * NCOLS;
    #pragma unroll
    for (int j = 0; j < 8; ++j) {
        const int idx = lane * 8 + j;
        atomicAdd(&o[idx], ex * z[idx]);
    }
}

// out[i] = unnorm[i] / max(denom,1-if-zero) + bias
__global__ void finalize_gat(float* __restrict__ out, const float* __restrict__ denom,
                             const float* __restrict__ b, int n_dst) {
    int i = blockIdx.x * blockDim.x + threadIdx.x;
    if (i >= n_dst * NCOLS) return;
    const int node = i >> 8, hd = i & 255, h = hd >> 6;
    float dn = denom[node * H_HEADS + h];
    dn = (dn > 0.f) ? dn : 1.0f;
    out[i] = out[i] / dn + b[hd];
}

// self-attention over S slots of 64-d, then mean over slots; one wave per node.
// slots 0..3 from f0, slots 4..7 from f1 (S==8 only).
template <int S>
__global__ void attn_pool_k(const float* __restrict__ f0, const float* __restrict__ f1,
                            float* __restrict__ out, int n, int out_stride, int out_off) {
    const int node = (blockIdx.x * blockDim.x + threadIdx.x) >> 5;
    const int lane = threadIdx.x & 31;
    if (node >= n) return;
    float fs[S][2];
    #pragma unroll
    for (int s = 0; s < S; ++s) {
        const float* base = (s < 4) ? (f0 + (size_t)node * NCOLS + s * DHEAD)
                                    : (f1 + (size_t)node * NCOLS + (s - 4) * DHEAD);
        fs[s][0] = base[lane * 2];
        fs[s][1] = base[lane * 2 + 1];
    }
    float sc[S][S];
    #pragma unroll
    for (int s = 0; s < S; ++s)
        #pragma unroll
        for (int t = 0; t < S; ++t) {
            float p = fs[s][0] * fs[t][0] + fs[s][1] * fs[t][1];
            #pragma unroll
            for (int off = 16; off > 0; off >>= 1) p += __shfl_xor(p, off, 32);
            sc[s][t] = p * 0.125f;              // / sqrt(64)
        }
    float acc0 = 0.f, acc1 = 0.f;
    #pragma unroll
    for (int s = 0; s < S; ++s) {
        float mx = sc[s][0];
        #pragma unroll
        for (int t = 1; t < S; ++t) mx = fmaxf(mx, sc[s][t]);
        float w[S], den = 0.f;
        #pragma unroll
        for (int t = 0; t < S; ++t) { w[t] = __expf(sc[s][t] - mx); den += w[t]; }
        const float inv = 1.0f / den;
        #pragma unroll
        for (int t = 0; t < S; ++t) {
            const float p = w[t] * inv;
            acc0 += p * fs[t][0];
            acc1 += p * fs[t][1];
        }
    }
    const float invS = 1.0f / (float)S;
    float* o = out + (size_t)node * out_stride + out_off + lane * 2;
    o[0] = acc0 * invS;
    o[1] = acc1 * invS;
}

// strided row copy: dst[r, dst_off + c] = src[r, c]
__global__ void copy_rows(const float* __restrict__ src, int src_stride,
                          float* __restrict__ dst, int dst_stride, int dst_off,
                          int rows, int cols) {
    int i = blockIdx.x * blockDim.x + threadIdx.x;
    if (i >= rows * cols) return;
    const int r = i / cols, c = i % cols;
    dst[(size_t)r * dst_stride + dst_off + c] = src[(size_t)r * src_stride + c];
}

// ---------------------------------------------------------------------------
static inline int ceil_div(int a, int b) { return (a + b - 1) / b; }

extern "C" void kernel_launch(void* const* d_in, const int* in_sizes, int n_in,
                              void* d_out, int out_size, void* d_ws, size_t ws_size,
                              hipStream_t stream) {
    const int N_SRC = 100000, N0 = 30000, N1 = 8000, E0 = 200000, E1 = 80000;

    const float* x_user = (const float*)d_in[0];
    const float* x_item = (const float*)d_in[1];
    const int* e_src[6] = { (const int*)d_in[2], (const int*)d_in[4], (const int*)d_in[6],
                            (const int*)d_in[8], (const int*)d_in[10], (const int*)d_in[12] };
    const int* e_dst[6] = { (const int*)d_in[3], (const int*)d_in[5], (const int*)d_in[7],
                            (const int*)d_in[9], (const int*)d_in[11], (const int*)d_in[13] };
    const float* W0  = (const float*)d_in[14];
    const float* al0 = (const float*)d_in[15];
    const float* ar0 = (const float*)d_in[16];
    const float* b0  = (const float*)d_in[17];
    const float* W1  = (const float*)d_in[18];
    const float* al1 = (const float*)d_in[19];
    const float* ar1 = (const float*)d_in[20];
    const float* b1  = (const float*)d_in[21];

    // workspace carve-up (floats)
    float* ws = (float*)d_ws;
    float*    z_src = ws;                                 // 100000*256
    float*    z_dst = z_src + (size_t)N_SRC * NCOLS;      // 30000*256
    float*    el    = z_dst + (size_t)N0 * NCOLS;         // 100000*4
    float*    er    = el + (size_t)N_SRC * H_HEADS;       // 30000*4
    unsigned* emax  = (unsigned*)(er + (size_t)N0 * H_HEADS);
    float*    denom = (float*)(emax + (size_t)N0 * H_HEADS);
    float*    outA  = denom + (size_t)N0 * H_HEADS;       // 30000*256 (out_ii)
    float*    outB  = outA + (size_t)N0 * NCOLS;          // (out_iu)
    float*    outC  = outB + (size_t)N0 * NCOLS;          // (out_ui)
    float*    xu1   = outC + (size_t)N0 * NCOLS;          // 30000*128
    float*    xi1   = xu1 + (size_t)N0 * 128;             // 30000*128

    const unsigned ENC_NEG_INF = 0x007FFFFFu;             // enc_ord(-inf)

    auto run_gat = [&](const float* hsrc, int n_src, int K,
                       const float* hdst, int n_dst,
                       const int* es, const int* ed, int E,
                       const float* W, const float* al, const float* ar, const float* b,
                       float* outbuf) {
        // z_src = hsrc @ W ; z_dst = hdst @ W
        {
            dim3 g(ceil_div(ceil_div(n_src, 16), 4), NCOLS / 16);
            gemm_wmma_f32<<<g, 128, 0, stream>>>(hsrc, W, z_src, n_src, K);
        }
        {
            dim3 g(ceil_div(ceil_div(n_dst, 16), 4), NCOLS / 16);
            gemm_wmma_f32<<<g, 128, 0, stream>>>(hdst, W, z_dst, n_dst, K);
        }
        head_dot<<<ceil_div(n_src * H_HEADS, 256), 256, 0, stream>>>(z_src, al, el, n_src);
        head_dot<<<ceil_div(n_dst * H_HEADS, 256), 256, 0, stream>>>(z_dst, ar, er, n_dst);
        fill_u32<<<ceil_div(n_dst * H_HEADS, 256), 256, 0, stream>>>(emax, ENC_NEG_INF, n_dst * H_HEADS);
        fill_f32<<<ceil_div(n_dst * H_HEADS, 256), 256, 0, stream>>>(denom, 0.f, n_dst * H_HEADS);
        fill_f32<<<ceil_div(n_dst * NCOLS, 256), 256, 0, stream>>>(outbuf, 0.f, n_dst * NCOLS);
        edge_max<<<ceil_div(E, 256), 256, 0, stream>>>(es, ed, el, er, emax, E);
        edge_scatter<<<ceil_div(E * 32, 256), 256, 0, stream>>>(es, ed, el, er, emax,
                                                               z_src, denom, outbuf, E);
        finalize_gat<<<ceil_div(n_dst * NCOLS, 256), 256, 0, stream>>>(outbuf, denom, b, n_dst);
    };

    // ---------------- layer 0 (input dim 64, n_dst = 30000) ----------------
    run_gat(x_item, N_SRC, 64, x_item, N0, e_src[0], e_dst[0], E0,
            W0 + 0 * 64 * NCOLS, al0 + 0 * NCOLS, ar0 + 0 * NCOLS, b0 + 0 * NCOLS, outA); // ii
    run_gat(x_item, N_SRC, 64, x_user, N0, e_src[1], e_dst[1], E0,
            W0 + 1 * 64 * NCOLS, al0 + 1 * NCOLS, ar0 + 1 * NCOLS, b0 + 1 * NCOLS, outB); // iu
    run_gat(x_user, N_SRC, 64, x_item, N0, e_src[2], e_dst[2], E0,
            W0 + 2 * 64 * NCOLS, al0 + 2 * NCOLS, ar0 + 2 * NCOLS, b0 + 2 * NCOLS, outC); // ui

    // xu1 = [x_user[:N0] | pool(out_iu)] ; xi1 = [x_item[:N0] | pool(out_ii ++ out_ui)]
    copy_rows<<<ceil_div(N0 * 64, 256), 256, 0, stream>>>(x_user, 64, xu1, 128, 0, N0, 64);
    copy_rows<<<ceil_div(N0 * 64, 256), 256, 0, stream>>>(x_item, 64, xi1, 128, 0, N0, 64);
    attn_pool_k<4><<<ceil_div(N0 * 32, 256), 256, 0, stream>>>(outB, nullptr, xu1, N0, 128, 64);
    attn_pool_k<8><<<ceil_div(N0 * 32, 256), 256, 0, stream>>>(outA, outC, xi1, N0, 128, 64);

    // ---------------- layer 1 (input dim 128, n_dst = 8000) ----------------
    run_gat(xi1, N0, 128, xi1, N1, e_src[3], e_dst[3], E1,
            W1 + 0 * 128 * NCOLS, al1 + 0 * NCOLS, ar1 + 0 * NCOLS, b1 + 0 * NCOLS, outA); // ii
    run_gat(xi1, N0, 128, xu1, N1, e_src[4], e_dst[4], E1,
            W1 + 1 * 128 * NCOLS, al1 + 1 * NCOLS, ar1 + 1 * NCOLS, b1 + 1 * NCOLS, outB); // iu
    run_gat(xu1, N0, 128, xi1, N1, e_src[5], e_dst[5], E1,
            W1 + 2 * 128 * NCOLS, al1 + 2 * NCOLS, ar1 + 2 * NCOLS, b1 + 2 * NCOLS, outC); // ui

    // final output: rows 0..7999 = [xu1[:8000] | pool(out_iu)] (192 cols)
    //               rows 8000..15999 = [xi1[:8000] | pool(out_ii ++ out_ui)]
    float* out = (float*)d_out;
    copy_rows<<<ceil_div(N1 * 128, 256), 256, 0, stream>>>(xu1, 128, out, 192, 0, N1, 128);
    copy_rows<<<ceil_div(N1 * 128, 256), 256, 0, stream>>>(xi1, 128, out + (size_t)N1 * 192, 192, 0, N1, 128);
    attn_pool_k<4><<<ceil_div(N1 * 32, 256), 256, 0, stream>>>(outB, nullptr, out, N1, 192, 128);
    attn_pool_k<8><<<ceil_div(N1 * 32, 256), 256, 0, stream>>>(outA, outC, out + (size_t)N1 * 192, N1, 192, 128);
}